// LayoutPenalty_69011534512261
// MI455X (gfx1250) — compile-verified
//
#include <hip/hip_runtime.h>
#include <stdint.h>

// ---------------------------------------------------------------------------
// LayoutPenalty for MI455X (gfx1250).
//
// Memory-bound streaming argmax-classify reduction:
//   587 MB of fp32 logits read once  ->  ~25 us floor at 23.3 TB/s.
// Path: double-buffered GLOBAL_LOAD_ASYNC_TO_LDS_B128 tile staging
// (perfectly coalesced 16B/lane global pattern, ASYNCcnt tracked),
// ds_load_b128 row reads from a 176B-padded LDS layout (conflict-free),
// 3-group max classification (special/letter/digit), exact integer-valued
// float atomics for a bit-deterministic result.
// ---------------------------------------------------------------------------

#define GRID_BLOCKS 2048
#define TPB 128
#define WAVE 32
#define B_SAMPLES 524288
#define L_POS 7
#define N_ROWS (B_SAMPLES * L_POS)               // 3,670,016
#define TILE_ROWS 128
#define N_TILES (N_ROWS / TILE_ROWS)             // 28,672
#define TILES_PER_BLOCK (N_TILES / GRID_BLOCKS)  // 14
#define ROW_BYTES 160                            // 40 * fp32
#define ROW_PAD 176                              // 11 * 16B: aligned + conflict-free
#define TILE_BYTES (TILE_ROWS * ROW_BYTES)       // 20,480
#define BUF_BYTES (TILE_ROWS * ROW_PAD)          // 22,528
#define CPT 10                                   // 16B chunks per thread per tile

static_assert(N_ROWS % TILE_ROWS == 0, "tile divisibility");
static_assert(N_TILES % GRID_BLOCKS == 0, "grid divisibility");
static_assert(TPB * CPT * 16 == TILE_BYTES, "copy coverage");

typedef const __attribute__((address_space(1))) unsigned char cgu8;
typedef __attribute__((address_space(3))) unsigned char lu8;

// The async builtin takes (int4 AS1* src, int4 AS3* dst, imm offset, imm cpol)
// per the clang diagnostic from the previous round.
typedef int v4i __attribute__((ext_vector_type(4)));
typedef __attribute__((address_space(1))) v4i* g_v4i_p;
typedef __attribute__((address_space(3))) v4i* l_v4i_p;

template <int N>
__device__ __forceinline__ void wait_asynccnt() {
#if __has_builtin(__builtin_amdgcn_s_wait_asynccnt)
  __builtin_amdgcn_s_wait_asynccnt(N);
#else
  asm volatile("s_wait_asynccnt %0" ::"n"(N) : "memory");
#endif
}

__device__ __forceinline__ void async_copy_16(cgu8* src, lu8* dst) {
#if __has_builtin(__builtin_amdgcn_global_load_async_to_lds_b128)
  __builtin_amdgcn_global_load_async_to_lds_b128((g_v4i_p)src, (l_v4i_p)dst, 0,
                                                 0);
#else
  asm volatile("global_load_async_to_lds_b128 %0, %1, off"
               ::"v"((unsigned)(unsigned long long)dst),
                 "v"((unsigned long long)src)
               : "memory");
#endif
}

// Copy one 128-row tile (20,480B) into LDS, rows padded to 176B.
// Global side: lane L, step k reads bytes [ (tid+128k)*16, +16 )  -> fully
// coalesced consecutive 16B chunks across the wave.
__device__ __forceinline__ void issue_tile(cgu8* gtile, lu8* lbuf, int tid) {
#pragma unroll
  for (int k = 0; k < CPT; ++k) {
    unsigned c = (unsigned)tid + (unsigned)(TPB * k);  // chunk id, < 1280
    unsigned row = (c * 52429u) >> 19;                 // c / 10 (exact for c < 81920)
    unsigned sub = c - row * 10u;                      // c % 10
    async_copy_16(gtile + (size_t)c * 16u, lbuf + row * ROW_PAD + sub * 16u);
  }
}

// Classify the argmax of a 40-float row; return 1 if it violates the mask.
// vocab: ids 0..3 special, 4..29 letters, 30..39 digits.
// First-index tie-breaking (jnp.argmax): special beats letter beats digit.
__device__ __forceinline__ int row_violation(const float* rowp, unsigned grow,
                                             const int* __restrict__ merc) {
  const float4* p = (const float4*)rowp;
  float4 a0 = p[0], a1 = p[1], a2 = p[2], a3 = p[3], a4 = p[4];
  float4 a5 = p[5], a6 = p[6], a7 = p[7], a8 = p[8], a9 = p[9];

  float mS = fmaxf(fmaxf(a0.x, a0.y), fmaxf(a0.z, a0.w));  // v0..v3

  float mL = fmaxf(fmaxf(a1.x, a1.y), fmaxf(a1.z, a1.w));  // v4..v7
  mL = fmaxf(mL, fmaxf(fmaxf(a2.x, a2.y), fmaxf(a2.z, a2.w)));
  mL = fmaxf(mL, fmaxf(fmaxf(a3.x, a3.y), fmaxf(a3.z, a3.w)));
  mL = fmaxf(mL, fmaxf(fmaxf(a4.x, a4.y), fmaxf(a4.z, a4.w)));
  mL = fmaxf(mL, fmaxf(fmaxf(a5.x, a5.y), fmaxf(a5.z, a5.w)));
  mL = fmaxf(mL, fmaxf(fmaxf(a6.x, a6.y), fmaxf(a6.z, a6.w)));  // ..v27
  mL = fmaxf(mL, fmaxf(a7.x, a7.y));                            // v28,v29

  float mD = fmaxf(a7.z, a7.w);                                 // v30,v31
  mD = fmaxf(mD, fmaxf(fmaxf(a8.x, a8.y), fmaxf(a8.z, a8.w)));
  mD = fmaxf(mD, fmaxf(fmaxf(a9.x, a9.y), fmaxf(a9.z, a9.w)));  // ..v39

  bool isLet = (mL > mS) & (mL >= mD);
  bool isDig = (mD > mS) & (mD > mL);

  unsigned samp = grow / 7u;
  unsigned pos = grow - samp * 7u;
  // Brazil LLLDDDD: letters at 0,1,2. Mercosul LLLDLDD: letters at 0,1,2,4.
  bool lmask = (pos < 3u) || ((pos == 4u) && (merc[samp] > 0));
  return lmask ? (int)isDig : (int)isLet;
}

__global__ void penalty_zero(float* __restrict__ out) { out[0] = 0.0f; }

__global__ __launch_bounds__(TPB) void penalty_main(
    const float* __restrict__ logits, const int* __restrict__ merc,
    float* __restrict__ out) {
  __shared__ __align__(16) unsigned char smem[2 * BUF_BYTES];
  const int tid = (int)threadIdx.x;
  const unsigned blk = blockIdx.x;

  cgu8* g = (cgu8*)logits;
  lu8* l = (lu8*)&smem[0];

  int cnt = 0;

  // Prologue: start tile 0 into buffer 0.
  issue_tile(g + (size_t)blk * TILE_BYTES, l, tid);

#pragma unroll
  for (int i = 0; i < TILES_PER_BLOCK; ++i) {
    const unsigned tcur = blk + (unsigned)i * GRID_BLOCKS;
    if (i + 1 < TILES_PER_BLOCK) {
      const unsigned tnxt = tcur + GRID_BLOCKS;
      issue_tile(g + (size_t)tnxt * TILE_BYTES,
                 l + (unsigned)(((i + 1) & 1) * BUF_BYTES), tid);
      wait_asynccnt<CPT>();  // 20 outstanding -> <=10 means current tile landed
    } else {
      wait_asynccnt<0>();
    }
    __syncthreads();  // all waves' async writes visible

    const float* rowp =
        (const float*)(smem + (i & 1) * BUF_BYTES + tid * ROW_PAD);
    cnt += row_violation(rowp, tcur * TILE_ROWS + (unsigned)tid, merc);

    __syncthreads();  // everyone done reading before this buffer is refilled
  }

  // Wave32 integer reduction, then one exact float atomic per wave.
#pragma unroll
  for (int off = WAVE / 2; off > 0; off >>= 1) cnt += __shfl_xor(cnt, off, WAVE);
  if ((tid & (WAVE - 1)) == 0) {
    // cnt <= 448; cnt * 2^-18 and all partial sums are exact in fp32:
    // deterministic regardless of atomic ordering, matches BETA*sum/B exactly.
    atomicAdd(out, (float)cnt * (2.0f / 524288.0f));
  }
}

extern "C" void kernel_launch(void* const* d_in, const int* in_sizes, int n_in,
                              void* d_out, int out_size, void* d_ws,
                              size_t ws_size, hipStream_t stream) {
  (void)in_sizes; (void)n_in; (void)out_size; (void)d_ws; (void)ws_size;
  const float* logits = (const float*)d_in[0];
  const int* merc = (const int*)d_in[1];
  float* out = (float*)d_out;

  penalty_zero<<<dim3(1), dim3(1), 0, stream>>>(out);
  penalty_main<<<dim3(GRID_BLOCKS), dim3(TPB), 0, stream>>>(logits, merc, out);
}